// GCN_TC_61229053772177
// MI455X (gfx1250) — compile-verified
//
#include <hip/hip_runtime.h>
#include <math.h>

typedef __attribute__((ext_vector_type(2))) float v2f;
typedef __attribute__((ext_vector_type(8))) float v8f;
typedef int vi4 __attribute__((vector_size(16)));

#define F_IN 1080
#define HIDC 128
#define OUTC 64
#define NGRAPH 256

#if defined(__has_builtin)
#if __has_builtin(__builtin_amdgcn_global_load_async_to_lds_b128)
#define HAVE_ASYNC_LDS 1
#endif
#endif
#ifndef HAVE_ASYNC_LDS
#define HAVE_ASYNC_LDS 0
#endif

// ---------------------------------------------------------------------------
// 1) Collapse the three (k=10, s=2) convs into one 64-tap filter + bias.
//    h[j] = sum_{a,b,c} w[a]w[b]w[c] x[8j + 4a + 2b + c] + cb*((Sw)^2 + Sw + 1)
// ---------------------------------------------------------------------------
__global__ void filter_kernel(const float* __restrict__ w,
                              const float* __restrict__ cb,
                              float* __restrict__ wc) {
    if (threadIdx.x != 0 || blockIdx.x != 0) return;
    float acc[64];
#pragma unroll
    for (int t = 0; t < 64; ++t) acc[t] = 0.f;
    float sw = 0.f;
    for (int a = 0; a < 10; ++a) sw += w[a];
    for (int a = 0; a < 10; ++a)
        for (int b = 0; b < 10; ++b) {
            float wab = w[a] * w[b];
            for (int c = 0; c < 10; ++c)
                acc[4 * a + 2 * b + c] += wab * w[c];
        }
#pragma unroll
    for (int t = 0; t < 64; ++t) wc[t] = acc[t];
    wc[64] = cb[0] * (sw * sw + sw + 1.0f);
}

// ---------------------------------------------------------------------------
// 2) Fused composite conv: one block per node, 128 threads = 128 outputs.
// ---------------------------------------------------------------------------
__global__ void __launch_bounds__(128) conv_kernel(const float* __restrict__ x,
                                                   const float* __restrict__ wc,
                                                   float* __restrict__ h) {
    __shared__ float row[F_IN];
    __shared__ float f[65];
    const int n = blockIdx.x;
    const float* xr = x + (long)n * F_IN;
    for (int i = threadIdx.x; i < F_IN; i += 128) row[i] = xr[i];
    if (threadIdx.x < 65) f[threadIdx.x] = wc[threadIdx.x];
    __syncthreads();
    const int j = threadIdx.x;
    const int base = 8 * j;
    float s = 0.f;
#pragma unroll
    for (int t = 0; t < 64; ++t) s = fmaf(f[t], row[base + t], s);
    h[(long)n * HIDC + j] = s + f[64];
}

// ---------------------------------------------------------------------------
// 3) GCN degree / symmetric normalization
// ---------------------------------------------------------------------------
__global__ void deg_init_kernel(float* __restrict__ deg, int N) {
    int i = blockIdx.x * blockDim.x + threadIdx.x;
    if (i < N) deg[i] = 1.0f;  // self-loop
}
__global__ void deg_accum_kernel(const int* __restrict__ dst, float* __restrict__ deg, int E) {
    int e = blockIdx.x * blockDim.x + threadIdx.x;
    if (e < E) atomicAdd(&deg[dst[e]], 1.0f);
}
__global__ void deg_finalize_kernel(float* __restrict__ deg, int N) {
    int i = blockIdx.x * blockDim.x + threadIdx.x;
    if (i < N) deg[i] = rsqrtf(fmaxf(deg[i], 1.0f));
}

// ---------------------------------------------------------------------------
// 4) Y = H @ W via V_WMMA_F32_16X16X4_F32 (full-fp32 WMMA).
//    512 threads = 16 waves; whole 128x128 W staged into LDS (64KB) with
//    GLOBAL_LOAD_ASYNC_TO_LDS_B128 (ASYNCcnt) when the toolchain exposes it.
//    Each wave: one 16-row M tile, 8 N tiles x 32 K-steps; A row lives in regs.
// ---------------------------------------------------------------------------
__global__ void __launch_bounds__(512) gemm128_wmma_kernel(const float* __restrict__ H,
                                                           const float* __restrict__ W,
                                                           float* __restrict__ Y, int N) {
    __shared__ float Ws[HIDC * HIDC];  // 64 KB

#if HAVE_ASYNC_LDS
    {
        // 512 lanes x 16B per sweep = 8KB; 8 sweeps for 64KB. VGPR-bypassing DMA.
        __attribute__((address_space(3))) float* ldsW =
            (__attribute__((address_space(3))) float*)Ws;
        __attribute__((address_space(1))) float* gW =
            (__attribute__((address_space(1))) float*)(float*)W;
        for (int i = threadIdx.x * 4; i < HIDC * HIDC; i += 512 * 4)
            __builtin_amdgcn_global_load_async_to_lds_b128(
                (__attribute__((address_space(1))) vi4*)(gW + i),
                (__attribute__((address_space(3))) vi4*)(ldsW + i), 0, 0);
        asm volatile("s_wait_asynccnt 0x0" ::: "memory");
    }
#else
    for (int i = threadIdx.x; i < HIDC * HIDC; i += 512) Ws[i] = W[i];
#endif
    __syncthreads();

    const int wave = threadIdx.x >> 5;
    const int lane = threadIdx.x & 31;
    const int m0 = (blockIdx.x * 16 + wave) * 16;
    if (m0 >= N) return;  // wave-uniform: EXEC stays all-ones for WMMA

    const int hh = lane >> 4;  // lane half
    const int nl = lane & 15;  // lane-in-half
    const float* hrow = H + (long)(m0 + nl) * HIDC;  // A: M = m0 + (lane&15)

    for (int n0 = 0; n0 < HIDC; n0 += 16) {
        v8f acc = {};
#pragma unroll
        for (int k0 = 0; k0 < HIDC; k0 += 4) {
            const int ka = k0 + 2 * hh;
            v2f a, b;
            a.x = hrow[ka];
            a.y = hrow[ka + 1];
            b.x = Ws[ka * HIDC + n0 + nl];        // K = k0 + 2*hh,   N = n0+nl
            b.y = Ws[(ka + 1) * HIDC + n0 + nl];  // K = k0 + 2*hh+1
            acc = __builtin_amdgcn_wmma_f32_16x16x4_f32(
                false, a, false, b, (short)0, acc, false, false);
        }
#pragma unroll
        for (int r = 0; r < 8; ++r)
            Y[(long)(m0 + r + 8 * hh) * HIDC + n0 + nl] = acc[r];
    }
}

// ---------------------------------------------------------------------------
// 5) Seed accumulator with the layer bias (segment_sum + b).
// ---------------------------------------------------------------------------
__global__ void bias_init_kernel(float* __restrict__ agg, const float* __restrict__ b, long total) {
    long i = (long)blockIdx.x * blockDim.x + threadIdx.x;
    if (i < total) agg[i] = b[i & (HIDC - 1)];
}

// ---------------------------------------------------------------------------
// 6) Edge scatter: one wave per edge (incl. self-loops), lane = 4 channels.
//    agg[dst] += xw[src] * dinv[src]*dinv[dst]   (f32 atomics, L2-resident)
// ---------------------------------------------------------------------------
__global__ void __launch_bounds__(256) scatter_kernel(const float* __restrict__ xw,
                                                      const float* __restrict__ dinv,
                                                      const int* __restrict__ src,
                                                      const int* __restrict__ dst,
                                                      int E, int N,
                                                      float* __restrict__ agg) {
    const int e = blockIdx.x * 8 + (threadIdx.x >> 5);
    const int lane = threadIdx.x & 31;
    if (e >= E + N) return;
    int s, d;
    if (e < E) { s = src[e]; d = dst[e]; } else { s = d = e - E; }
    const float nrm = dinv[s] * dinv[d];
    const float4 v = ((const float4*)(xw + (long)s * HIDC))[lane];  // b128 load
    float* ad = agg + (long)d * HIDC + lane * 4;
    atomicAdd(ad + 0, v.x * nrm);
    atomicAdd(ad + 1, v.y * nrm);
    atomicAdd(ad + 2, v.z * nrm);
    atomicAdd(ad + 3, v.w * nrm);
}

__global__ void relu_kernel(float* __restrict__ a, long total) {
    long i = (long)blockIdx.x * blockDim.x + threadIdx.x;
    if (i < total) a[i] = fmaxf(a[i], 0.0f);
}

// ---------------------------------------------------------------------------
// 7) Global mean pool: one wave per node, 4 channels/lane, f32 atomics.
// ---------------------------------------------------------------------------
__global__ void pool_init_kernel(float* __restrict__ pooled, float* __restrict__ cnt) {
    int i = blockIdx.x * blockDim.x + threadIdx.x;
    if (i < NGRAPH * HIDC) pooled[i] = 0.f;
    if (i < NGRAPH) cnt[i] = 0.f;
}
__global__ void __launch_bounds__(256) pool_kernel(const float* __restrict__ h,
                                                   const int* __restrict__ batch,
                                                   float* __restrict__ pooled,
                                                   float* __restrict__ cnt, int N) {
    const int n = blockIdx.x * 8 + (threadIdx.x >> 5);
    const int lane = threadIdx.x & 31;
    if (n >= N) return;
    const int g = batch[n];
    const float4 v = ((const float4*)(h + (long)n * HIDC))[lane];
    float* pd = pooled + (long)g * HIDC + lane * 4;
    atomicAdd(pd + 0, v.x);
    atomicAdd(pd + 1, v.y);
    atomicAdd(pd + 2, v.z);
    atomicAdd(pd + 3, v.w);
    if (lane == 0) atomicAdd(&cnt[g], 1.0f);
}

// ---------------------------------------------------------------------------
// 8) Tiny head MLP: one thread per graph (256 total, ~2 MFLOP).
// ---------------------------------------------------------------------------
__global__ void __launch_bounds__(256) mlp_kernel(const float* __restrict__ pooled,
                                                  const float* __restrict__ cnt,
                                                  const float* __restrict__ l1w,
                                                  const float* __restrict__ l1b,
                                                  const float* __restrict__ l2w,
                                                  const float* __restrict__ l2b,
                                                  float* __restrict__ out) {
    const int g = threadIdx.x;
    if (blockIdx.x != 0 || g >= NGRAPH) return;
    const float inv = 1.0f / fmaxf(cnt[g], 1.0f);
    const float* p = pooled + (long)g * HIDC;
    float o = 0.f;
    for (int j = 0; j < OUTC; ++j) {
        float a = 0.f;
        for (int c = 0; c < HIDC; ++c) a = fmaf(p[c], l1w[c * OUTC + j], a);
        a = fmaxf(a * inv + l1b[j], 0.0f);
        o = fmaf(a, l2w[j], o);
    }
    o += l2b[0];
    out[g] = 1.0f / (1.0f + expf(-o));
}

// ---------------------------------------------------------------------------
extern "C" void kernel_launch(void* const* d_in, const int* in_sizes, int n_in,
                              void* d_out, int out_size, void* d_ws, size_t ws_size,
                              hipStream_t stream) {
    const float* x      = (const float*)d_in[0];
    const int*   eidx   = (const int*)d_in[1];
    const int*   batch  = (const int*)d_in[2];
    const float* conv_w = (const float*)d_in[3];
    const float* conv_b = (const float*)d_in[4];
    const float* Wl[3]  = { (const float*)d_in[5], (const float*)d_in[7], (const float*)d_in[9] };
    const float* bl[3]  = { (const float*)d_in[6], (const float*)d_in[8], (const float*)d_in[10] };
    const float* l1w    = (const float*)d_in[11];
    const float* l1b    = (const float*)d_in[12];
    const float* l2w    = (const float*)d_in[13];
    const float* l2b    = (const float*)d_in[14];
    float* out = (float*)d_out;

    const int N = in_sizes[0] / F_IN;
    const int E = in_sizes[1] / 2;
    const int* src = eidx;
    const int* dst = eidx + E;

    // Workspace layout (floats): ~52 MB total — edge/feature working set is
    // L2-resident on MI455X (192 MB L2), HBM mostly sees x once (216 MB).
    float* ws     = (float*)d_ws;
    float* wcomp  = ws;                        // 65  (pad to 256)
    float* dinv   = ws + 256;                  // N
    float* buf0   = dinv + N;                  // N*128 : h / agg
    float* buf1   = buf0 + (long)N * HIDC;     // N*128 : h@W
    float* pooled = buf1 + (long)N * HIDC;     // 256*128
    float* cnt    = pooled + NGRAPH * HIDC;    // 256

    const long tot = (long)N * HIDC;
    const int blk256 = 256;

    // conv filter + composite conv
    filter_kernel<<<1, 64, 0, stream>>>(conv_w, conv_b, wcomp);
    conv_kernel<<<N, 128, 0, stream>>>(x, wcomp, buf0);

    // degree -> dinv
    deg_init_kernel<<<(N + 255) / 256, blk256, 0, stream>>>(dinv, N);
    deg_accum_kernel<<<(E + 255) / 256, blk256, 0, stream>>>(dst, dinv, E);
    deg_finalize_kernel<<<(N + 255) / 256, blk256, 0, stream>>>(dinv, N);

    // three GCN layers
    const int gemm_grid = (N + 255) / 256;  // 16 waves * 16 rows / block
    const int scat_grid = (E + N + 7) / 8;  // one wave per edge
    for (int l = 0; l < 3; ++l) {
        gemm128_wmma_kernel<<<gemm_grid, 512, 0, stream>>>(buf0, Wl[l], buf1, N);
        bias_init_kernel<<<(int)((tot + 255) / 256), blk256, 0, stream>>>(buf0, bl[l], tot);
        scatter_kernel<<<scat_grid, 256, 0, stream>>>(buf1, dinv, src, dst, E, N, buf0);
        if (l < 2)
            relu_kernel<<<(int)((tot + 255) / 256), blk256, 0, stream>>>(buf0, tot);
    }

    // pooling + head
    pool_init_kernel<<<(NGRAPH * HIDC + 255) / 256, blk256, 0, stream>>>(pooled, cnt);
    pool_kernel<<<(N + 7) / 8, 256, 0, stream>>>(buf0, batch, pooled, cnt, N);
    mlp_kernel<<<1, 256, 0, stream>>>(pooled, cnt, l1w, l1b, l2w, l2b, out);
}